// TransE_46677704573832
// MI455X (gfx1250) — compile-verified
//
#include <hip/hip_runtime.h>

typedef __attribute__((ext_vector_type(2))) float v2f;
typedef __attribute__((ext_vector_type(8))) float v8f;

#define DIM 128
#define MARGIN 1.0f

// One wave (32 lanes) processes 16 consecutive triples (= 8 samples).
// Lane L supplies the f32 WMMA A-matrix slice for row (L&15), K-offset (L>>4)*2.
// acc = Gram(diff, diff); diagonal = squared L2 distances of the 16 triples.
__global__ __launch_bounds__(256) void transe_partial_kernel(
    const float* __restrict__ ent, const float* __restrict__ rel,
    const int* __restrict__ tb, float* __restrict__ partial, int n_triples)
{
    __shared__ float wsum[8];
    const int lane = threadIdx.x & 31;
    const int wave = threadIdx.x >> 5;
    const long long wid  = (long long)blockIdx.x * 8 + wave;
    const long long base = wid * 16;                 // first triple of this wave

    long long tri_ll = base + (lane & 15);
    if (tri_ll > n_triples - 1) tri_ll = n_triples - 1;   // clamp (masked later)
    const int tri = (int)tri_ll;

    const int ih = tb[3 * tri + 0];
    const int ir = tb[3 * tri + 1];
    const int it = tb[3 * tri + 2];

    const int o = (lane >> 4) * 2;                   // K sub-offset within chunk
    const float* ph = ent + (size_t)ih * DIM + o;
    const float* pr = rel + (size_t)ir * DIM + o;
    const float* pt = ent + (size_t)it * DIM + o;

    v8f acc = {};
    #pragma unroll 8
    for (int c = 0; c < 32; ++c) {                   // 32 K-chunks of 4 cover DIM=128
        v2f h = *(const v2f*)(ph + 4 * c);
        v2f r = *(const v2f*)(pr + 4 * c);
        v2f t = *(const v2f*)(pt + 4 * c);
        v2f d = h + r - t;
        // D = A*B + C with A == B == diff slice -> Gram matrix accumulate
        acc = __builtin_amdgcn_wmma_f32_16x16x4_f32(
            /*neg_a=*/false, d, /*neg_b=*/false, d,
            /*c_mod=*/(short)0, acc, /*reuse_a=*/false, /*reuse_b=*/false);
    }

    // Diagonal extraction per the documented C/D layout:
    //   lanes 0-15 : VGPR v holds (M=v,   N=lane)    -> diag i=0..7  at lane i,  acc[i]
    //   lanes 16-31: VGPR v holds (M=v+8, N=lane-16) -> diag i=8..15 at lane i+16, acc[i-8]
    const bool valid = (lane < 8) || (lane >= 24);
    const int  vidx  = (lane < 8) ? lane : (lane - 24);   // acc element holding my diag
    const int  myTri = (lane < 8) ? lane : (lane - 16);   // triple index within the 16

    float dsq = 0.0f;
    #pragma unroll
    for (int i = 0; i < 8; ++i) dsq = (vidx == i) ? acc[i] : dsq;

    float dist  = valid ? __builtin_sqrtf(dsq) : 0.0f;
    float dnext = __shfl_down(dist, 1, 32);          // partner (corrupt) distance

    // Even triple = true triple; its corrupt partner sits one lane up.
    float term = 0.0f;
    if (valid && ((myTri & 1) == 0) && (base + myTri + 1 < (long long)n_triples))
        term = fmaxf(0.0f, MARGIN + dist - dnext);

    // Wave reduction (wave32)
    #pragma unroll
    for (int off = 16; off > 0; off >>= 1) term += __shfl_xor(term, off, 32);

    if (lane == 0) wsum[wave] = term;
    __syncthreads();
    if (threadIdx.x == 0) {
        float s = 0.0f;
        #pragma unroll
        for (int i = 0; i < 8; ++i) s += wsum[i];
        partial[blockIdx.x] = s;
    }
}

// Deterministic fixed-order final reduction of block partials.
__global__ __launch_bounds__(256) void transe_reduce_kernel(
    const float* __restrict__ partial, int n, float* __restrict__ out)
{
    __shared__ float sm[256];
    float s = 0.0f;
    for (int i = threadIdx.x; i < n; i += 256) s += partial[i];
    sm[threadIdx.x] = s;
    __syncthreads();
    #pragma unroll
    for (int step = 128; step > 0; step >>= 1) {
        if ((int)threadIdx.x < step) sm[threadIdx.x] += sm[threadIdx.x + step];
        __syncthreads();
    }
    if (threadIdx.x == 0) out[0] = sm[0];
}

extern "C" void kernel_launch(void* const* d_in, const int* in_sizes, int n_in,
                              void* d_out, int out_size, void* d_ws, size_t ws_size,
                              hipStream_t stream) {
    const float* ent = (const float*)d_in[0];   // (E, 128) f32
    const float* rel = (const float*)d_in[1];   // (R, 128) f32
    const int*   tb  = (const int*)d_in[2];     // (B, 2, 3) int32, flat

    const int n_triples = in_sizes[2] / 3;            // B*2 = 262144
    const int waves     = (n_triples + 15) / 16;      // 16 triples per wave
    const int blocks    = (waves + 7) / 8;            // 8 waves per block

    float* partial = (float*)d_ws;                    // blocks floats of scratch

    transe_partial_kernel<<<blocks, 256, 0, stream>>>(ent, rel, tb, partial, n_triples);
    transe_reduce_kernel<<<1, 256, 0, stream>>>(partial, blocks, (float*)d_out);
}